// GraphConv_24197845745955
// MI455X (gfx1250) — compile-verified
//
#include <hip/hip_runtime.h>
#include <hip/hip_bf16.h>

typedef __attribute__((ext_vector_type(2))) float v2f;
typedef __attribute__((ext_vector_type(8))) float v8f;

// ---------------------------------------------------------------------------
// 1) deg[i] = 1.0  (self-loop contribution)
// ---------------------------------------------------------------------------
__global__ void gcn_init_deg(float* __restrict__ deg, int n) {
    int i = blockIdx.x * blockDim.x + threadIdx.x;
    if (i < n) deg[i] = 1.0f;
}

// ---------------------------------------------------------------------------
// 2) deg[dst[e]] += 1  over all edges
// ---------------------------------------------------------------------------
__global__ void gcn_count_deg(const long long* __restrict__ dst,
                              float* __restrict__ deg, int e) {
    int i = blockIdx.x * blockDim.x + threadIdx.x;
    if (i < e) atomicAdd(&deg[(int)dst[i]], 1.0f);
}

// ---------------------------------------------------------------------------
// 3) dinv[i] = rsqrt(deg[i])   (deg >= 1 always, so no zero guard needed)
// ---------------------------------------------------------------------------
__global__ void gcn_rsqrt(const float* __restrict__ deg,
                          float* __restrict__ dinv, int n) {
    int i = blockIdx.x * blockDim.x + threadIdx.x;
    if (i < n) dinv[i] = rsqrtf(deg[i]);
}

// ---------------------------------------------------------------------------
// 4) h = x @ W  via V_WMMA_F32_16X16X4_F32 (fp32 matrix pipe, wave32)
//    block = 256 threads = 8 waves; each block owns a 16-row stripe of x
//    (staged in LDS, reused by all 8 waves); wave w computes N-tile w.
// ---------------------------------------------------------------------------
__global__ void __launch_bounds__(256)
gcn_gemm_wmma(const float* __restrict__ x, const float* __restrict__ W,
              float* __restrict__ h) {
    __shared__ float As[16 * 128];          // 8 KB x-tile

    const int m0 = blockIdx.x * 16;

    // cooperative load of the 16x128 A tile (512 float4 by 256 threads)
    const float4* xv = (const float4*)(x + (size_t)m0 * 128);
    float4* asv = (float4*)As;
    for (int i = threadIdx.x; i < 16 * 128 / 4; i += 256) asv[i] = xv[i];
    __syncthreads();

    const int wave = threadIdx.x >> 5;      // 0..7 -> N tile
    const int lane = threadIdx.x & 31;
    const int n0   = wave * 16;
    const int mrow = lane & 15;             // M (for A) / N (for B) within tile
    const int kh   = (lane >> 4) * 2;       // lanes 16-31 hold K+2,K+3

    v8f c = {};
    #pragma unroll
    for (int k = 0; k < 128; k += 4) {
        v2f a, b;
        a.x = As[mrow * 128 + k + kh];
        a.y = As[mrow * 128 + k + kh + 1];
        b.x = W[(size_t)(k + kh)     * 128 + n0 + mrow];
        b.y = W[(size_t)(k + kh + 1) * 128 + n0 + mrow];
        // D = A(16x4,f32) * B(4x16,f32) + C(16x16,f32)
        c = __builtin_amdgcn_wmma_f32_16x16x4_f32(
                /*neg_a=*/false, a, /*neg_b=*/false, b,
                /*c_mod=*/(short)0, c, /*reuse_a=*/false, /*reuse_b=*/false);
    }

    // C/D layout: VGPR v -> row m0 + v + 8*(lane>=16), col n0 + (lane&15)
    const int rbase = m0 + (lane >> 4) * 8;
    const int col   = n0 + (lane & 15);
    #pragma unroll
    for (int v = 0; v < 8; v++)
        h[(size_t)(rbase + v) * 128 + col] = c[v];
}

// ---------------------------------------------------------------------------
// 5) out[i,:] = h[i,:] * dinv[i]^2 + b[:]   (self-loop term + bias; also
//    serves as the initialization of the scatter accumulator -> no zero pass)
//    one thread per float4: N*32 threads
// ---------------------------------------------------------------------------
__global__ void gcn_init_out(const float* __restrict__ h,
                             const float* __restrict__ dinv,
                             const float* __restrict__ b,
                             float* __restrict__ out, int n) {
    int t = blockIdx.x * blockDim.x + threadIdx.x;
    if (t >= n * 32) return;
    int row = t >> 5;
    int q   = t & 31;                        // float4 index within row
    float w = dinv[row];
    w = w * w;
    float4 hv = ((const float4*)(h + (size_t)row * 128))[q];
    float4 bv = ((const float4*)b)[q];
    float4 o;
    o.x = hv.x * w + bv.x;
    o.y = hv.y * w + bv.y;
    o.z = hv.z * w + bv.z;
    o.w = hv.w * w + bv.w;
    ((float4*)(out + (size_t)row * 128))[q] = o;
}

// ---------------------------------------------------------------------------
// 6) edge scatter: out[dst] += h[src] * (dinv[src]*dinv[dst])
//    one wave (32 lanes) per edge; each lane owns a float4 (covers 128 f32).
//    h and out are L2-resident (51 MB each vs 192 MB L2) so the gathers and
//    global_atomic_add_f32 resolve in L2.
// ---------------------------------------------------------------------------
__global__ void __launch_bounds__(256)
gcn_scatter(const long long* __restrict__ src, const long long* __restrict__ dst,
            const float* __restrict__ h, const float* __restrict__ dinv,
            float* __restrict__ out, int e) {
    int edge = blockIdx.x * 8 + (threadIdx.x >> 5);
    if (edge >= e) return;
    int lane = threadIdx.x & 31;
    int s = (int)src[edge];
    int d = (int)dst[edge];
    float w = dinv[s] * dinv[d];
    float4 v = ((const float4*)(h + (size_t)s * 128))[lane];
    float* o = out + (size_t)d * 128 + lane * 4;
    atomicAdd(o + 0, v.x * w);
    atomicAdd(o + 1, v.y * w);
    atomicAdd(o + 2, v.z * w);
    atomicAdd(o + 3, v.w * w);
}

// ---------------------------------------------------------------------------
extern "C" void kernel_launch(void* const* d_in, const int* in_sizes, int n_in,
                              void* d_out, int out_size, void* d_ws, size_t ws_size,
                              hipStream_t stream) {
    const float*     x  = (const float*)d_in[0];
    const long long* ei = (const long long*)d_in[1];   // int64 [2, E]
    const float*     W  = (const float*)d_in[2];
    const float*     b  = (const float*)d_in[3];
    float* out = (float*)d_out;

    const int N = in_sizes[0] / 128;
    const int E = in_sizes[1] / 2;
    const long long* src = ei;
    const long long* dst = ei + E;

    // workspace layout: deg[N] | dinv[N] | h[N*128]
    float* deg  = (float*)d_ws;
    float* dinv = deg + N;
    float* h    = dinv + N;

    const int B256 = 256;

    // 1) self-loop degree
    gcn_init_deg<<<(N + B256 - 1) / B256, B256, 0, stream>>>(deg, N);
    // 2) in-degree count
    gcn_count_deg<<<(E + B256 - 1) / B256, B256, 0, stream>>>(dst, deg, E);
    // 3) dinv = rsqrt(deg)
    gcn_rsqrt<<<(N + B256 - 1) / B256, B256, 0, stream>>>(deg, dinv, N);
    // 4) h = x @ W   (N divisible by 16: 100000/16 = 6250)
    gcn_gemm_wmma<<<N / 16, 256, 0, stream>>>(x, W, h);
    // 5) out = h*dinv^2 + b  (self-loop + bias, init accumulator)
    gcn_init_out<<<(N * 32 + B256 - 1) / B256, B256, 0, stream>>>(h, dinv, b, out, N);
    // 6) edge scatter-add
    gcn_scatter<<<(E + 7) / 8, 256, 0, stream>>>(src, dst, h, dinv, out, E);
}